// CGIterator_45483703665084
// MI455X (gfx1250) — compile-verified
//
#include <hip/hip_runtime.h>

// ---------------------------------------------------------------------------
// CG tensor-product iteration for MI455X (gfx1250, wave32, WMMA).
//   [A] cg_con_kernel<l1,l2,L> : one launch per CG contribution; fully
//       compile-time shapes -> unrolled FMA, static LDS for the CG slice.
//       Writes its channel stripe of cat (ws) + per-channel sum(x^2) atomics.
//   [B] rms1_kernel            : inv_rms[i] = rsqrt(mean)
//   [C] wmma_gemm_kernel<SIZE,KOUT>: one wave per 16-row stripe, holding all
//       KOUT/16 accumulator tiles; K-loop step 4 with V_WMMA_F32_16X16X4_F32.
//       A-fragment loaded once per K-step and reused across NT WMMAs.
//       Also accumulates global sum(y^2) (wave shuffle + one atomic).
//   [D] finalize_kernel        : y *= rsqrt(mean(y^2)); y += feats[L] if S==1
// Workspace: stats (few KB) + one reusable cat buffer (max 896*N floats,
// ~72MB for N=20000 -> resident in the 192MB L2).
// ---------------------------------------------------------------------------

typedef float v2f __attribute__((ext_vector_type(2)));
typedef float v8f __attribute__((ext_vector_type(8)));

__global__ void zero_kernel(float* p, int n) {
  int i = blockIdx.x * blockDim.x + threadIdx.x;
  if (i < n) p[i] = 0.0f;
}

// ---- [A] one CG contribution, compile-time shaped --------------------------
template <int l1, int l2, int LL>
__global__ void cg_con_kernel(const float* __restrict__ F1,
                              const float* __restrict__ F2,
                              const float* __restrict__ cg_all,
                              float* __restrict__ cat,
                              float* __restrict__ chan_ss,
                              int size, int base, int N) {
  constexpr int n1 = 2 * l1 + 1, n2 = 2 * l2 + 1, nm = 2 * LL + 1;
  constexpr int K1 = (l1 == 0) ? 128 : (l1 == 1) ? 64 : (l1 == 2) ? 32 : 16;
  constexpr int K2 = (l2 == 0) ? 128 : (l2 == 1) ? 64 : (l2 == 2) ? 32 : 16;
  constexpr int KMIN = K1 < K2 ? K1 : K2;
  constexpr int CGN = n1 * n2 * nm;
  constexpr int CGBASE = ((l1 * 4 + l2) * 4 + LL) * 343;  // cg_all (4,4,4,7,7,7)

  __shared__ float scg[CGN];
  for (int e = threadIdx.x; e < CGN; e += blockDim.x) {
    int a = e / (n2 * nm), b = (e / nm) % n2, mm = e % nm;
    scg[e] = cg_all[CGBASE + a * 49 + b * 7 + mm];
  }
  __syncthreads();

  int k = threadIdx.x;
  if (k >= KMIN) return;

  float lss = 0.0f;
  for (int n = blockIdx.x; n < N; n += gridDim.x) {
    float t1[n1], t2[n2];
#pragma unroll
    for (int a = 0; a < n1; ++a) t1[a] = F1[(n * n1 + a) * K1 + k];
#pragma unroll
    for (int b = 0; b < n2; ++b) t2[b] = F2[(n * n2 + b) * K2 + k];
#pragma unroll
    for (int mm = 0; mm < nm; ++mm) {
      float s = 0.0f;
#pragma unroll
      for (int a = 0; a < n1; ++a)
#pragma unroll
        for (int b = 0; b < n2; ++b)
          s += t1[a] * t2[b] * scg[(a * n2 + b) * nm + mm];
      cat[(size_t)(n * nm + mm) * size + base + k] = s;
      lss += s * s;
    }
  }
  atomicAdd(&chan_ss[base + k], lss);
}

// ---- [B] per-channel inverse RMS -------------------------------------------
__global__ void rms1_kernel(const float* __restrict__ chan_ss,
                            float* __restrict__ inv_rms, int size, float count) {
  int i = blockIdx.x * blockDim.x + threadIdx.x;
  if (i < size) inv_rms[i] = rsqrtf(chan_ss[i] / count);
}

// ---- [C] channel-mix GEMM via V_WMMA_F32_16X16X4_F32 -----------------------
// One wave32 per 16-row stripe of the (M x KOUT) output; NT = KOUT/16
// accumulator tiles live in registers so the A fragment is loaded once per
// K-step. fp32 A 16x4 layout: lanes 0-15 rows (K cols 0,1), lanes 16-31 (2,3).
template <int SIZE, int KOUT>
__global__ void __launch_bounds__(32)
wmma_gemm_kernel(const float* __restrict__ cat, const float* __restrict__ W,
                 const float* __restrict__ inv_rms, float* __restrict__ y,
                 float* __restrict__ scal_ss, int M) {
  constexpr int NT = KOUT / 16;
  int lane = threadIdx.x;
  int half = lane >> 4;
  int lr   = lane & 15;
  int r0   = blockIdx.x * 16;
  int arow = r0 + lr;
  int arowc = arow < M ? arow : M - 1;  // clamp (M is a multiple of 16 anyway)
  const float* arowp = cat + (size_t)arowc * SIZE;

  __builtin_prefetch(arowp, 0, 1);  // global_prefetch_b8: warm the A-row stream

  v8f acc[NT];
#pragma unroll
  for (int t = 0; t < NT; ++t) acc[t] = (v8f){0, 0, 0, 0, 0, 0, 0, 0};

#if __has_builtin(__builtin_amdgcn_wmma_f32_16x16x4_f32)
  for (int kk = 0; kk < SIZE; kk += 4) {
    int ac = kk + 2 * half;
    v2f ir = *(const v2f*)(inv_rms + ac);
    v2f av = *(const v2f*)(arowp + ac);
    v2f a;
    a.x = av.x * ir.x;
    a.y = av.y * ir.y;
#pragma unroll
    for (int t = 0; t < NT; ++t) {
      v2f b;
      b.x = W[ac * KOUT + t * 16 + lr];
      b.y = W[(ac + 1) * KOUT + t * 16 + lr];
      acc[t] = __builtin_amdgcn_wmma_f32_16x16x4_f32(
          /*neg_a=*/false, a, /*neg_b=*/false, b,
          /*c_mod=*/(short)0, acc[t], /*reuse_a=*/false, /*reuse_b=*/false);
    }
  }
#else
  for (int kk = 0; kk < SIZE; ++kk) {
    float ir = inv_rms[kk];
#pragma unroll
    for (int t = 0; t < NT; ++t) {
      float bv = W[kk * KOUT + t * 16 + lr];
#pragma unroll
      for (int g = 0; g < 8; ++g) {
        int row = r0 + half * 8 + g;
        int rc  = row < M ? row : M - 1;
        acc[t][g] += cat[(size_t)rc * SIZE + kk] * ir * bv;
      }
    }
  }
#endif

  float ss = 0.0f;
#pragma unroll
  for (int t = 0; t < NT; ++t) {
#pragma unroll
    for (int g = 0; g < 8; ++g) {
      int row = r0 + half * 8 + g;  // C/D: VGPR g -> M=g (lo half), M=g+8 (hi)
      float v = acc[t][g];
      if (row < M) y[(size_t)row * KOUT + t * 16 + lr] = v;
      ss += (row < M) ? v * v : 0.0f;
    }
  }
  for (int off = 16; off > 0; off >>= 1) ss += __shfl_down(ss, off, 32);
  if (lane == 0) atomicAdd(scal_ss, ss);
}

// ---- [D] global RMS scale + residual, in place in d_out --------------------
__global__ void finalize_kernel(float* __restrict__ out,
                                const float* __restrict__ scal_ss,
                                const float* __restrict__ resid, int S,
                                size_t count) {
  float scale = rsqrtf(*scal_ss / (float)count);
  for (size_t idx = blockIdx.x * (size_t)blockDim.x + threadIdx.x; idx < count;
       idx += (size_t)gridDim.x * blockDim.x) {
    float v = out[idx] * scale;
    if (S == 1) v += resid[idx];
    out[idx] = v;
  }
}

// ---------------------------------------------------------------------------
// Host-side dispatch tables.
// ---------------------------------------------------------------------------
#define MAXCON 6

// All 32 (l1,l2,L) contributions actually launched (group 2 truncated to the
// 176 channels consistent with W_2_1's shape).
#define CON_LIST \
  X(0,0,0) X(1,1,0) X(2,2,0) X(3,3,0) \
  X(0,1,1) X(1,0,1) X(1,2,1) X(2,1,1) X(2,3,1) X(3,2,1) \
  X(0,2,2) X(1,1,2) X(1,3,2) X(2,0,2) X(2,2,2) \
  X(0,3,3) X(1,2,3) X(2,1,3) X(2,3,3) X(3,0,3) X(3,2,3) \
  X(1,1,1) X(2,2,1) X(3,3,1) \
  X(1,2,2) X(2,1,2) X(2,3,2) X(3,2,2) \
  X(1,3,3) X(2,2,3) X(3,1,3) X(3,3,3)

static inline void launch_con(int l1, int l2, int L, const float* const feat[4],
                              const float* cg_all, float* cat, float* chss,
                              int size, int base, int N, hipStream_t stream) {
  const int key = l1 * 100 + l2 * 10 + L;
  switch (key) {
#define X(A, B, C)                                                          \
  case A * 100 + B * 10 + C: {                                              \
    constexpr int KA = (A == 0) ? 128 : (A == 1) ? 64 : (A == 2) ? 32 : 16; \
    constexpr int KB = (B == 0) ? 128 : (B == 1) ? 64 : (B == 2) ? 32 : 16; \
    constexpr int KM = KA < KB ? KA : KB;                                   \
    constexpr int BD = KM < 32 ? 32 : KM;                                   \
    cg_con_kernel<A, B, C><<<dim3(256), dim3(BD), 0, stream>>>(             \
        feat[A], feat[B], cg_all, cat, chss, size, base, N);                \
  } break;
    CON_LIST
#undef X
    default: break;
  }
}

extern "C" void kernel_launch(void* const* d_in, const int* in_sizes, int n_in,
                              void* d_out, int out_size, void* d_ws, size_t ws_size,
                              hipStream_t stream) {
  const float* feat[4] = {(const float*)d_in[0], (const float*)d_in[1],
                          (const float*)d_in[2], (const float*)d_in[3]};
  const float* cg_all = (const float*)d_in[4];
  const float* W[7];
  for (int g = 0; g < 7; ++g) W[g] = (const float*)d_in[5 + g];
  float* out = (float*)d_out;
  float* ws  = (float*)d_ws;

  const int N = in_sizes[0] / 128;  // feat_0 is (N,1,128)

  // Target groups in ORDER: (0,1),(1,1),(2,1),(3,1),(1,-1),(2,-1),(3,-1).
  static const int Ltab[7] = {0, 1, 2, 3, 1, 2, 3};
  static const int Stab[7] = {1, 1, 1, 1, -1, -1, -1};
  static const int SZ[7]   = {240, 224, 176, 128, 112, 96, 80};
  static const int NC[7]   = {4, 6, 5, 6, 3, 4, 4};
  static const int C_L1[7][MAXCON] = {
      {0, 1, 2, 3, 0, 0}, {0, 1, 1, 2, 2, 3}, {0, 1, 1, 2, 2, 0},
      {0, 1, 2, 2, 3, 3}, {1, 2, 3, 0, 0, 0}, {1, 2, 2, 3, 0, 0},
      {1, 2, 3, 3, 0, 0}};
  static const int C_L2[7][MAXCON] = {
      {0, 1, 2, 3, 0, 0}, {1, 0, 2, 1, 3, 2}, {2, 1, 3, 0, 2, 0},
      {3, 2, 1, 3, 0, 2}, {1, 2, 3, 0, 0, 0}, {2, 1, 3, 2, 0, 0},
      {3, 2, 1, 3, 0, 0}};
  static const int C_BASE[7][MAXCON] = {
      {0, 128, 192, 224, 0, 0}, {0, 64, 128, 160, 192, 208},
      {0, 32, 96, 112, 144, 0}, {0, 16, 48, 80, 96, 112},
      {0, 64, 96, 0, 0, 0},     {0, 32, 64, 80, 0, 0},
      {0, 16, 48, 64, 0, 0}};
  static const int CH_OFF[7]    = {0, 240, 464, 640, 768, 880, 976};  // cum SZ
  static const int OUT_OFF_U[7] = {0, 128, 320, 480, 592, 784, 944};  // * N
  static const int KOUT[7]      = {128, 64, 32, 16, 64, 32, 16};

  // Workspace layout (floats):
  //   [0,1056)    per-channel sum(x^2)  (7 groups packed at CH_OFF)
  //   [1056,2112) per-channel inv_rms
  //   [2112,2119) per-group scalar sum(y^2)
  //   [4096, ...) reusable cat buffer (max (2L+1)*size = 896 floats per node)
  float* chan_ss = ws;
  float* inv_rms = ws + 1056;
  float* scal_ss = ws + 2112;
  float* cat     = ws + 4096;
  size_t need = (4096 + (size_t)896 * (size_t)N) * sizeof(float);
  if (ws_size < need) return;  // requires ~72MB of scratch

  zero_kernel<<<(2119 + 255) / 256, 256, 0, stream>>>(ws, 2119);

  for (int g = 0; g < 7; ++g) {
    const int L = Ltab[g], S = Stab[g], size = SZ[g];
    const int nm = 2 * L + 1;
    const int M  = N * nm;
    float* yout  = out + (size_t)OUT_OFF_U[g] * N;
    float* chss  = chan_ss + CH_OFF[g];

    for (int c = 0; c < NC[g]; ++c)
      launch_con(C_L1[g][c], C_L2[g][c], L, feat, cg_all, cat, chss, size,
                 C_BASE[g][c], N, stream);

    rms1_kernel<<<(size + 255) / 256, 256, 0, stream>>>(
        chss, inv_rms + CH_OFF[g], size, (float)((double)N * nm));

    dim3 gc((M + 15) / 16);
    switch (g) {
      case 0: wmma_gemm_kernel<240, 128><<<gc, 32, 0, stream>>>(
                  cat, W[g], inv_rms + CH_OFF[g], yout, scal_ss + g, M); break;
      case 1: wmma_gemm_kernel<224, 64><<<gc, 32, 0, stream>>>(
                  cat, W[g], inv_rms + CH_OFF[g], yout, scal_ss + g, M); break;
      case 2: wmma_gemm_kernel<176, 32><<<gc, 32, 0, stream>>>(
                  cat, W[g], inv_rms + CH_OFF[g], yout, scal_ss + g, M); break;
      case 3: wmma_gemm_kernel<128, 16><<<gc, 32, 0, stream>>>(
                  cat, W[g], inv_rms + CH_OFF[g], yout, scal_ss + g, M); break;
      case 4: wmma_gemm_kernel<112, 64><<<gc, 32, 0, stream>>>(
                  cat, W[g], inv_rms + CH_OFF[g], yout, scal_ss + g, M); break;
      case 5: wmma_gemm_kernel<96, 32><<<gc, 32, 0, stream>>>(
                  cat, W[g], inv_rms + CH_OFF[g], yout, scal_ss + g, M); break;
      case 6: wmma_gemm_kernel<80, 16><<<gc, 32, 0, stream>>>(
                  cat, W[g], inv_rms + CH_OFF[g], yout, scal_ss + g, M); break;
    }

    size_t count = (size_t)M * (size_t)KOUT[g];
    finalize_kernel<<<1024, 256, 0, stream>>>(yout, scal_ss + g, feat[L], S,
                                              count);
  }
}